// NNDistance_52785148068622
// MI455X (gfx1250) — compile-verified
//
#include <hip/hip_runtime.h>
#include <hip/hip_bf16.h>

typedef __attribute__((ext_vector_type(2))) float v2f;
typedef __attribute__((ext_vector_type(8))) float v8f;

#define BATCH 8
#define NPTS  4096
#define CHUNK 512     // reference points staged in LDS per block iteration
#define WAVES 8       // waves per 256-thread block

// One direction of NN search: for each query row i, min_j ||q_i - r_j||^2.
// dist(i,j) computed entirely inside V_WMMA_F32_16X16X4_F32:
//   A row i = {-2x,-2y,-2z, 1}, B col j = {x,y,z,|r_j|^2}, C = |q_i|^2.
__global__ __launch_bounds__(256) void nnd_dir(
    const float* __restrict__ q,      // (BATCH, NPTS, 3) query points
    const float* __restrict__ r,      // (BATCH, NPTS, 3) reference points
    float*       __restrict__ odist,  // (BATCH, NPTS)
    int*         __restrict__ oidx)   // (BATCH, NPTS)
{
    __shared__ float4 spts[CHUNK];          // staged {x, y, z, |r|^2}
    __shared__ float  red_d[WAVES][16][16]; // per-row candidate dists
    __shared__ int    red_j[WAVES][16][16]; // per-row candidate indices

    const int  lane     = threadIdx.x & 31;
    const int  wv       = threadIdx.x >> 5;
    const int  gw       = blockIdx.x * WAVES + wv;  // global wave id
    const int  b        = gw >> 8;                  // NPTS/16 = 256 row tiles/batch
    const int  row_base = (gw & 255) << 4;
    const int  lrow     = lane & 15;
    const bool hi       = lane >= 16;

    // ---- A fragment (16x4 f32): lanes 0-15 hold {K0,K1}, lanes 16-31 {K2,K3}
    const float* qp = q + ((size_t)b * NPTS + row_base + lrow) * 3;
    const float qx = qp[0], qy = qp[1], qz = qp[2];
    const float n1 = qx * qx + qy * qy + qz * qz;
    v2f a;
    a.x = hi ? (-2.0f * qz) : (-2.0f * qx);
    a.y = hi ? 1.0f         : (-2.0f * qy);

    // ---- C fragment: C[M][N] = |q_M|^2 broadcast along N.
    // Row M's norm lives in lane M (and lane M+16); slot v maps to M = v + 8*hi.
    v8f c;
#pragma unroll
    for (int v = 0; v < 8; ++v)
        c[v] = __shfl(n1, v + (hi ? 8 : 0), 32);

    float bestd[8];
    int   bestj[8];
#pragma unroll
    for (int v = 0; v < 8; ++v) { bestd[v] = 3.0e38f; bestj[v] = 0; }

    const float* rb = r + (size_t)b * NPTS * 3;

    for (int chunk = 0; chunk < NPTS; chunk += CHUNK) {
        __syncthreads();  // protect spts from previous iteration's readers
        for (int t = threadIdx.x; t < CHUNK; t += 256) {
            const float* pp = rb + (size_t)(chunk + t) * 3;
            const float x = pp[0], y = pp[1], z = pp[2];
            spts[t] = make_float4(x, y, z, x * x + y * y + z * z);
        }
        __syncthreads();

#pragma unroll 4
        for (int tt = 0; tt < CHUNK; tt += 16) {
            // B fragment (4x16 f32): lanes 0-15 read {x,y}, lanes 16-31 {z,|r|^2}
            const float2* sp =
                (const float2*)((const char*)&spts[tt + lrow] + (hi ? 8 : 0));
            const float2 pb = *sp;
            v2f bf; bf.x = pb.x; bf.y = pb.y;

            // D[M][N] = |q_M|^2 - 2 q_M . r_N + |r_N|^2  (full squared distance)
            v8f d = __builtin_amdgcn_wmma_f32_16x16x4_f32(
                false, a, false, bf, (short)0, c, false, false);

            const int gj = chunk + tt + lrow;
#pragma unroll
            for (int v = 0; v < 8; ++v) {
                if (d[v] < bestd[v]) { bestd[v] = d[v]; bestj[v] = gj; }
            }
        }
    }

    // ---- Cross-lane argmin: deposit (row, col-residue) partials to LDS,
    // then 16 lanes/wave scan 16 candidates each (lowest-index tie-break).
#pragma unroll
    for (int v = 0; v < 8; ++v) {
        const int rr = v + (hi ? 8 : 0);
        red_d[wv][rr][lrow] = bestd[v];
        red_j[wv][rr][lrow] = bestj[v];
    }
    __syncthreads();
    if (lane < 16) {
        float bd = red_d[wv][lane][0];
        int   bj = red_j[wv][lane][0];
#pragma unroll
        for (int k = 1; k < 16; ++k) {
            const float dk = red_d[wv][lane][k];
            const int   jk = red_j[wv][lane][k];
            if (dk < bd || (dk == bd && jk < bj)) { bd = dk; bj = jk; }
        }
        const size_t o = (size_t)b * NPTS + row_base + lane;
        odist[o] = bd;
        oidx[o]  = bj;
    }
}

extern "C" void kernel_launch(void* const* d_in, const int* in_sizes, int n_in,
                              void* d_out, int out_size, void* d_ws, size_t ws_size,
                              hipStream_t stream) {
    (void)in_sizes; (void)n_in; (void)d_ws; (void)ws_size; (void)out_size;
    const float* xyz1 = (const float*)d_in[0];
    const float* xyz2 = (const float*)d_in[1];

    float* out   = (float*)d_out;
    float* dist1 = out;
    int*   idx1  = (int*)(out + (size_t)BATCH * NPTS);
    float* dist2 = out + 2 * (size_t)BATCH * NPTS;
    int*   idx2  = (int*)(out + 3 * (size_t)BATCH * NPTS);

    const int grid = BATCH * (NPTS / 16) / WAVES;  // 256 blocks of 8 waves
    // Direction 1: for each point in xyz1, nearest in xyz2
    nnd_dir<<<grid, 256, 0, stream>>>(xyz1, xyz2, dist1, idx1);
    // Direction 2: symmetric, swap inputs
    nnd_dir<<<grid, 256, 0, stream>>>(xyz2, xyz1, dist2, idx2);
}